// GNN_28509992911125
// MI455X (gfx1250) — compile-verified
//
#include <hip/hip_runtime.h>

// ---------------------------------------------------------------------------
// GIN forward on gfx1250 (MI455X).
//   per layer: agg = h + scatter_add(h[src] -> dst)          (atomic, L2-bound)
//              z1  = relu(bn(agg @ W1 + b1))                 (WMMA f32 16x16x4)
//              h   = bn(z1 @ W2 + b2) (+relu except last)    (WMMA f32 16x16x4)
//   pooled = segment_sum(h, batch); out = leaky_relu(pooled @ Wp + bp, 0.1)
//
// All activations live in workspace with row stride LDH=112 (7 col tiles),
// pads zero-filled, and weights are pre-packed transposed+padded, so the
// WMMA k-loop has NO guards: pure global_load_b64 + v_wmma_f32_16x16x4_f32.
// ---------------------------------------------------------------------------

#define NN  50000
#define EE  800000
#define GG  512
#define LDH 112
#define BN_INV 0.99999500003749981f   // 1/sqrt(1+1e-5)

typedef float v2f __attribute__((ext_vector_type(2)));
typedef float v8f __attribute__((ext_vector_type(8)));

__global__ void zero_f32(float* __restrict__ p, int n) {
  int i = blockIdx.x * blockDim.x + threadIdx.x;
  int st = gridDim.x * blockDim.x;
  for (; i < n; i += st) p[i] = 0.0f;
}

// dst[node][0:ldout) = (c < C) ? src[node][c] : 0   (wave per node)
__global__ void init_pad(const float* __restrict__ src, int ldin,
                         float* __restrict__ dst, int ldout, int n, int C) {
  int warp  = (blockIdx.x * blockDim.x + threadIdx.x) >> 5;
  int lane  = threadIdx.x & 31;
  int nwarp = (gridDim.x * blockDim.x) >> 5;
  for (int node = warp; node < n; node += nwarp) {
    const float* s = src + (long)node * ldin;
    float*       d = dst + (long)node * ldout;
    for (int c = lane; c < ldout; c += 32)
      d[c] = (c < C) ? s[c] : 0.0f;
  }
}

// WT[c][k] = (c<Cout && k<K) ? W[k][c] : 0   packed to Cpad x Kpad
__global__ void pack_wt(const float* __restrict__ W, float* __restrict__ WT,
                        int K, int Cout, int Kpad, int Cpad) {
  int i = blockIdx.x * blockDim.x + threadIdx.x;
  int st = gridDim.x * blockDim.x;
  int total = Kpad * Cpad;
  for (; i < total; i += st) {
    int c = i / Kpad;
    int k = i - c * Kpad;
    WT[(long)c * Kpad + k] = (c < Cout && k < K) ? W[(long)k * Cout + c] : 0.0f;
  }
}

// one wave per edge; lanes stride channels (coalesced 128B strips)
__global__ void edge_scatter(const float* __restrict__ h, int ldin,
                             const int* __restrict__ src, const int* __restrict__ dst,
                             float* __restrict__ agg, int ldout, int E, int C) {
  int warp  = (blockIdx.x * blockDim.x + threadIdx.x) >> 5;
  int lane  = threadIdx.x & 31;
  int nwarp = (gridDim.x * blockDim.x) >> 5;
  for (int e = warp; e < E; e += nwarp) {
    int s = src[e], d = dst[e];
    const float* hs = h + (long)s * ldin;
    float*       ad = agg + (long)d * ldout;
    for (int c = lane; c < C; c += 32)
      atomicAdd(&ad[c], hs[c]);
  }
}

// one wave per node; scatter into per-graph pooled rows
__global__ void pool_scatter(const float* __restrict__ h, const int* __restrict__ batch,
                             float* __restrict__ pooled, int n, int C) {
  int warp  = (blockIdx.x * blockDim.x + threadIdx.x) >> 5;
  int lane  = threadIdx.x & 31;
  int nwarp = (gridDim.x * blockDim.x) >> 5;
  for (int node = warp; node < n; node += nwarp) {
    int g = batch[node];
    const float* hn = h + (long)node * LDH;
    float*       pg = pooled + (long)g * LDH;
    for (int c = lane; c < C; c += 32)
      atomicAdd(&pg[c], hn[c]);
  }
}

// ---------------------------------------------------------------------------
// WMMA GEMM: out = epilogue(A[M x Kpad] @ WT^T + bias)
//   A: row stride lda (pads zero).  WT: [Cpad x Kpad] transposed, zero-padded.
//   One wave owns a 16-row strip and NCT col tiles (A-fragment reuse).
//   Unguarded v2f loads in the k-loop; EXEC stays all-ones around wmma.
//   epilogue mode 0: y = gamma*BN_INV*(acc+b)+beta, relu
//                 1: same, no relu
//                 2: y = acc + b, leaky_relu(0.1)
//   Fragment layouts per CDNA5 ISA 7.12.2 (h = lane>>4, m = lane&15):
//     A 16x4 : a = A[row m][kb+2h .. +1]
//     B 4x16 : b = W[kb+2h .. +1][col m]  == WT[col m][kb+2h .. +1]
//     C/D    : col = m, VGPR r holds row r + 8h
// ---------------------------------------------------------------------------
template <int NCT, bool PADOUT>
__global__ __launch_bounds__(256) void wmma_gemm(
    const float* __restrict__ A, int lda,
    const float* __restrict__ WT, int ldb,
    const float* __restrict__ bias, const float* __restrict__ gamma,
    const float* __restrict__ beta,
    float* __restrict__ out, int ldc,
    int rowTiles, int Kpad, int Cout, int mode) {
  int wave = (blockIdx.x * blockDim.x + threadIdx.x) >> 5;
  if (wave >= rowTiles) return;               // wave-uniform: EXEC stays all-ones
  int lane = threadIdx.x & 31;
  int m    = lane & 15;
  int half = lane >> 4;
  long rowBase = (long)wave * 16;

  const v8f vzero = {0.f, 0.f, 0.f, 0.f, 0.f, 0.f, 0.f, 0.f};
  v8f acc[NCT];
#pragma unroll
  for (int ct = 0; ct < NCT; ++ct) acc[ct] = vzero;

  const float* arow = A + (rowBase + m) * (long)lda + 2 * half;
  const float* wcol = WT + (long)m * ldb + 2 * half;

  for (int kb = 0; kb < Kpad; kb += 4) {
    v2f a = *(const v2f*)(arow + kb);
#pragma unroll
    for (int ct = 0; ct < NCT; ++ct) {
      v2f b = *(const v2f*)(wcol + (long)ct * 16 * ldb + kb);
      acc[ct] = __builtin_amdgcn_wmma_f32_16x16x4_f32(
          false, a, false, b, (short)0, acc[ct], false, false);
    }
  }

#pragma unroll
  for (int ct = 0; ct < NCT; ++ct) {
    int col = ct * 16 + m;
    if (PADOUT) {
      float bi = 0.f, ga = 0.f, be = 0.f;
      bool cok = (col < Cout);
      if (cok) {                      // loads under guard; store unguarded
        bi = bias[col];
        ga = (mode == 2) ? 1.0f : gamma[col];
        be = (mode == 2) ? 0.0f : beta[col];
      }
#pragma unroll
      for (int r = 0; r < 8; ++r) {
        float y = 0.0f;
        if (cok) {
          float v = acc[ct][r];
          if (mode == 2) {
            y = v + bi;
            y = (y < 0.0f) ? 0.1f * y : y;
          } else {
            y = ga * (BN_INV * (v + bi)) + be;
            if (mode == 0) y = fmaxf(y, 0.0f);
          }
        }
        long row = rowBase + r + 8 * half;
        out[row * (long)ldc + col] = y;     // pads get 0 (next GEMM reads them)
      }
    } else {
      if (col < Cout) {
        float bi = bias[col];
#pragma unroll
        for (int r = 0; r < 8; ++r) {
          float y = acc[ct][r] + bi;
          if (mode == 2) y = (y < 0.0f) ? 0.1f * y : y;
          long row = rowBase + r + 8 * half;
          out[row * (long)ldc + col] = y;
        }
      }
    }
  }
}

extern "C" void kernel_launch(void* const* d_in, const int* in_sizes, int n_in,
                              void* d_out, int out_size, void* d_ws, size_t ws_size,
                              hipStream_t stream) {
  const float* x     = (const float*)d_in[0];
  const int*   eidx  = (const int*)d_in[1];
  const int*   esrc  = eidx;          // edge_index[0]
  const int*   edst  = eidx + EE;     // edge_index[1]
  const int*   batch = (const int*)d_in[2];

  // params flattened: 4 layers x (W1,b1,g1,be1,W2,b2,g2,be2), then Wp, bp
  const float* P[40];
  for (int i = 3; i < n_in && i < 40; ++i) P[i] = (const float*)d_in[i];
  const float* Wp = P[35];
  const float* bp = P[36];

  float* agg    = (float*)d_ws;                   // N*LDH
  float* z1     = agg + (long)NN * LDH;           // N*LDH
  float* h      = z1  + (long)NN * LDH;           // N*LDH
  float* pooled = h   + (long)NN * LDH;           // G*LDH
  float* wbuf   = pooled + (long)GG * LDH;        // packed weights
  float* outp   = (float*)d_out;                  // 512*12

  const int IC[4] = {33, 100, 100, 100};
  const int OC[4] = {100, 100, 100, 100};
  const int TPB  = 256;
  const int SBLK = 2048;
  const int rowTiles = NN / 16;                   // 3125 exact
  const int gemmBlocks = (rowTiles + 7) / 8;

  // ---- pack all weights (transposed, zero-padded) ----
  float* w1t[4];
  float* w2t[4];
  float* cur = wbuf;
  for (int L = 0; L < 4; ++L) {
    int ic = IC[L], oc = OC[L];
    int kp = (ic + 3) & ~3;                       // 36 or 100
    int cp1 = ((ic + 15) / 16) * 16;              // 48 or 112
    int cp2 = ((oc + 15) / 16) * 16;              // 112
    w1t[L] = cur; cur += (long)cp1 * kp;
    w2t[L] = cur; cur += (long)cp2 * kp;
    pack_wt<<<64, TPB, 0, stream>>>(P[3 + 8*L + 0], w1t[L], ic, ic, kp, cp1);
    pack_wt<<<64, TPB, 0, stream>>>(P[3 + 8*L + 4], w2t[L], ic, oc, kp, cp2);
  }
  float* wpt = cur;                                // 16 x 100
  pack_wt<<<64, TPB, 0, stream>>>(Wp, wpt, 100, 12, 100, 16);

  // ---- layers ----
  for (int L = 0; L < 4; ++L) {
    const float* hin = (L == 0) ? x : h;
    int ldin = (L == 0) ? 33 : LDH;
    int ic = IC[L], oc = OC[L];
    int kp = (ic + 3) & ~3;
    int base = 3 + 8 * L;
    const float *b1 = P[base+1], *g1 = P[base+2], *be1 = P[base+3];
    const float *b2 = P[base+5], *g2 = P[base+6], *be2 = P[base+7];

    // agg = h (self term, eps=0), padded+zero-filled, then += edges
    init_pad<<<SBLK, TPB, 0, stream>>>(hin, ldin, agg, LDH, NN, ic);
    edge_scatter<<<SBLK, TPB, 0, stream>>>(hin, ldin, esrc, edst, agg, LDH, EE, ic);

    // z1 = relu(bn(agg @ W1 + b1))
    if (ic == 33)
      wmma_gemm<3, true><<<gemmBlocks, TPB, 0, stream>>>(
          agg, LDH, w1t[L], kp, b1, g1, be1, z1, LDH, rowTiles, kp, ic, 0);
    else
      wmma_gemm<7, true><<<gemmBlocks, TPB, 0, stream>>>(
          agg, LDH, w1t[L], kp, b1, g1, be1, z1, LDH, rowTiles, kp, ic, 0);

    // h = bn(z1 @ W2 + b2) (+relu except last layer)
    int mode2 = (L < 3) ? 0 : 1;
    wmma_gemm<7, true><<<gemmBlocks, TPB, 0, stream>>>(
        z1, LDH, w2t[L], kp, b2, g2, be2, h, LDH, rowTiles, kp, oc, mode2);
  }

  // ---- global_add_pool + projection ----
  zero_f32<<<64, TPB, 0, stream>>>(pooled, GG * LDH);
  pool_scatter<<<SBLK, TPB, 0, stream>>>(h, batch, pooled, NN, 100);

  int poolTiles = GG / 16;                        // 32
  wmma_gemm<1, false><<<(poolTiles + 7) / 8, TPB, 0, stream>>>(
      pooled, LDH, wpt, 100, bp, nullptr, nullptr, outp, 12,
      poolTiles, 100, 12, 2);
}